// MultiLayerScorer_70248485093977
// MI455X (gfx1250) — compile-verified
//
#include <hip/hip_runtime.h>

typedef __attribute__((ext_vector_type(2))) float v2f;
typedef __attribute__((ext_vector_type(8))) float v8f;

#define BATCH 16
#define SRCN  100
#define TGTN  100
#define KDIM  512   // query/key feature dims (M == N == 512)
#define HDIM  256
#define STILE 4     // s-rows per scoring block (100 = 25 * 4)

// fast tanh: prefer the gfx1250 v_tanh_f32 trans op; fall back to exp2+rcp
__device__ __forceinline__ float fast_tanh(float x) {
#if __has_builtin(__builtin_amdgcn_tanhf)
  return __builtin_amdgcn_tanhf(x);
#else
  // tanh(x) = 1 - 2/(e^{2x}+1);  e^{2x} = exp2(x * 2*log2(e))
  float e = __builtin_amdgcn_exp2f(x * 2.8853900817779268f);
  return 1.0f - 2.0f * __builtin_amdgcn_rcpf(e + 1.0f);
#endif
}

// -----------------------------------------------------------------------------
// Kernel 1: both projections via V_WMMA_F32_16X16X4_F32, register-blocked 4x1:
// each wave owns a 64x16 output panel (4 stacked 16x16 tiles), so the B
// fragment is loaded once per K-step and reused by 4 WMMAs (4 independent
// accumulator chains keep the matrix pipe busy).
//   hk  = keys  @ Wk         stored row-major (1600, 256)
//   hqT = query @ Wq + b1    stored transposed (16, 256, 100)
// 1600 rows = 25 tiles of 64; 256 cols = 16 tiles; x2 matrices = 800 waves.
// -----------------------------------------------------------------------------
__global__ __launch_bounds__(256) void proj_wmma_kernel(
    const float* __restrict__ query,   // (1600, 512) flat
    const float* __restrict__ keys,    // (1600, 512) flat
    const float* __restrict__ Wk,      // (512, 256)
    const float* __restrict__ Wq,      // (512, 256)
    const float* __restrict__ b1,      // (256)
    float* __restrict__ hk,            // (1600, 256)
    float* __restrict__ hqT)           // (16, 256, 100)
{
  const int wave = threadIdx.x >> 5;
  const int lane = threadIdx.x & 31;
  int tile = blockIdx.x * 8 + wave;           // 0..799
  const int isQuery = (tile >= 400) ? 1 : 0;
  if (isQuery) tile -= 400;
  const int rowTile = tile >> 4;              // 0..24  (64 rows each)
  const int colTile = tile & 15;              // 0..15  (16 cols each)

  const float* __restrict__ A = isQuery ? query : keys;
  const float* __restrict__ W = isQuery ? Wq : Wk;

  const int mn   = lane & 15;                 // M for A-lane, N for B-lane
  const int half = lane >> 4;                 // 0 or 1
  const int koff = half * 2;                  // K sub-offset per ISA layout

  const float* __restrict__ Arow0 = A + (size_t)(rowTile * 64 + mn) * KDIM;
  const float* __restrict__ Arow1 = Arow0 + (size_t)16 * KDIM;
  const float* __restrict__ Arow2 = Arow0 + (size_t)32 * KDIM;
  const float* __restrict__ Arow3 = Arow0 + (size_t)48 * KDIM;
  const float* __restrict__ Wcol  = W + (colTile * 16 + mn);

  v8f acc0 = {}, acc1 = {}, acc2 = {}, acc3 = {};
  #pragma unroll 2
  for (int k = 0; k < KDIM; k += 4) {
    const int kk = k + koff;
    v2f b;                                     // B 4x16 fragment, shared by 4 tiles
    b.x = Wcol[(size_t)kk * HDIM];
    b.y = Wcol[(size_t)(kk + 1) * HDIM];
    v2f a0, a1, a2, a3;                        // A 16x4 fragments (b64 loads)
    a0.x = Arow0[kk]; a0.y = Arow0[kk + 1];
    a1.x = Arow1[kk]; a1.y = Arow1[kk + 1];
    a2.x = Arow2[kk]; a2.y = Arow2[kk + 1];
    a3.x = Arow3[kk]; a3.y = Arow3[kk + 1];
    acc0 = __builtin_amdgcn_wmma_f32_16x16x4_f32(false, a0, false, b, (short)0, acc0, false, false);
    acc1 = __builtin_amdgcn_wmma_f32_16x16x4_f32(false, a1, false, b, (short)0, acc1, false, false);
    acc2 = __builtin_amdgcn_wmma_f32_16x16x4_f32(false, a2, false, b, (short)0, acc2, false, false);
    acc3 = __builtin_amdgcn_wmma_f32_16x16x4_f32(false, a3, false, b, (short)0, acc3, false, false);
  }

  const int col = colTile * 16 + mn;
  v8f accs[4] = {acc0, acc1, acc2, acc3};
  if (!isQuery) {
    #pragma unroll
    for (int sub = 0; sub < 4; ++sub) {
      #pragma unroll
      for (int v = 0; v < 8; ++v) {
        const int r = rowTile * 64 + sub * 16 + v + half * 8;   // global key row
        hk[(size_t)r * HDIM + col] = accs[sub][v];
      }
    }
  } else {
    const float bias = b1[col];
    #pragma unroll
    for (int sub = 0; sub < 4; ++sub) {
      #pragma unroll
      for (int v = 0; v < 8; ++v) {
        const int r  = rowTile * 64 + sub * 16 + v + half * 8;  // global query row
        const int bb = r / TGTN;
        const int tt = r - bb * TGTN;
        hqT[((size_t)bb * HDIM + col) * TGTN + tt] = accs[sub][v] + bias;
      }
    }
  }
}

// -----------------------------------------------------------------------------
// Kernel 2: fused broadcast-add + tanh + <.,W2> reduction + output tanh.
// One block per (b, sTile of 4); thread t owns score[b, s0..s0+3, t]. Each
// hqT element loaded once feeds 4 tanh+FMA (4 hk rows broadcast from LDS),
// cutting L2 traffic 4x and giving 16 independent trans chains per unrolled
// step. hqT reads are coalesced across lanes (t contiguous). The 41M tanh
// (one per (b,s,t,h)) is the irreducible cost.
// -----------------------------------------------------------------------------
__global__ __launch_bounds__(128) void score_kernel(
    const float* __restrict__ hk,      // (1600, 256)
    const float* __restrict__ hqT,     // (16, 256, 100), b1 already folded in
    const float* __restrict__ W2,      // (256)
    const float* __restrict__ b2,      // (1)
    float* __restrict__ out)           // (16, 100, 100) = (b, s, t)
{
  __shared__ float hk_s[STILE][HDIM];  // 4 consecutive hk rows
  __shared__ float w2_s[HDIM];

  const int bst = blockIdx.x;          // b * 25 + sTile
  const int b   = bst / 25;
  const int s0  = (bst - b * 25) * STILE;
  const int tid = threadIdx.x;

  // rows (b*100+s0 .. +3) are contiguous: 1024 floats, 8 per thread
  const float* __restrict__ hkBase = hk + (size_t)(b * SRCN + s0) * HDIM;
  #pragma unroll
  for (int i = 0; i < STILE * HDIM / 128; ++i)
    ((float*)hk_s)[tid + i * 128] = hkBase[tid + i * 128];
  w2_s[tid]       = W2[tid];
  w2_s[tid + 128] = W2[tid + 128];
  __syncthreads();

  if (tid < TGTN) {
    const float* __restrict__ q = hqT + (size_t)b * HDIM * TGTN + tid;
    float acc0 = 0.0f, acc1 = 0.0f, acc2 = 0.0f, acc3 = 0.0f;
    #pragma unroll 4
    for (int h = 0; h < HDIM; ++h) {
      const float qv = q[(size_t)h * TGTN];   // coalesced, reused 4x
      const float w  = w2_s[h];               // LDS broadcast
      acc0 += fast_tanh(hk_s[0][h] + qv) * w;
      acc1 += fast_tanh(hk_s[1][h] + qv) * w;
      acc2 += fast_tanh(hk_s[2][h] + qv) * w;
      acc3 += fast_tanh(hk_s[3][h] + qv) * w;
    }
    const float b2v = b2[0];
    float* __restrict__ o = out + (size_t)(b * SRCN + s0) * TGTN + tid;
    o[0 * TGTN] = fast_tanh(acc0 + b2v);
    o[1 * TGTN] = fast_tanh(acc1 + b2v);
    o[2 * TGTN] = fast_tanh(acc2 + b2v);
    o[3 * TGTN] = fast_tanh(acc3 + b2v);
  }
}

extern "C" void kernel_launch(void* const* d_in, const int* in_sizes, int n_in,
                              void* d_out, int out_size, void* d_ws, size_t ws_size,
                              hipStream_t stream) {
  (void)in_sizes; (void)n_in; (void)out_size; (void)ws_size;

  const float* query = (const float*)d_in[0];  // (16,100,512)
  const float* keys  = (const float*)d_in[1];  // (16,100,512)
  const float* Wk    = (const float*)d_in[2];  // (512,256)
  const float* Wq    = (const float*)d_in[3];  // (512,256)
  const float* b1    = (const float*)d_in[4];  // (256)
  const float* W2    = (const float*)d_in[5];  // (256,1)
  const float* b2    = (const float*)d_in[6];  // (1)
  float* out = (float*)d_out;

  float* hk  = (float*)d_ws;                               // 1600*256 floats
  float* hqT = hk + (size_t)BATCH * SRCN * HDIM;           // 16*256*100 floats

  // 800 panel-waves (25 row-panels x 16 col-tiles x 2 matrices), 8 per block
  proj_wmma_kernel<<<100, 256, 0, stream>>>(query, keys, Wk, Wq, b1, hk, hqT);
  // one block per (b, 4-row s tile): 16 * 25 = 400 blocks
  score_kernel<<<BATCH * (SRCN / STILE), 128, 0, stream>>>(hk, hqT, W2, b2, out);
}